// SemanticPatternLossCalculator_46523085750355
// MI455X (gfx1250) — compile-verified
//
#include <hip/hip_runtime.h>
#include <hip/hip_bf16.h>
#include <math.h>

#define NUM_BINS 32
#define NUM_PAT 16
#define BATCH 16384
#define DIM 768
#define NCLS (NUM_BINS * NUM_PAT)

typedef __attribute__((ext_vector_type(2))) float v2f;
typedef __attribute__((ext_vector_type(8))) float v8f;

// ---------------- wave32 reduction helpers ----------------
__device__ __forceinline__ float wave_sum(float v) {
#pragma unroll
  for (int off = 16; off > 0; off >>= 1) v += __shfl_xor(v, off, 32);
  return v;
}
__device__ __forceinline__ float wave_max(float v) {
#pragma unroll
  for (int off = 16; off > 0; off >>= 1) v = fmaxf(v, __shfl_xor(v, off, 32));
  return v;
}
// argmax with first-occurrence tie-break (smaller index wins on equal value)
__device__ __forceinline__ void wave_argmax(float& v, int& i) {
#pragma unroll
  for (int off = 16; off > 0; off >>= 1) {
    float ov = __shfl_xor(v, off, 32);
    int   oi = __shfl_xor(i, off, 32);
    if (ov > v || (ov == v && oi < i)) { v = ov; i = oi; }
  }
}

// ---------------- K0: zero counters + accumulators ----------------
__global__ void k_init(int* counts, float* accum) {
  int t = threadIdx.x;
  if (t < NUM_BINS) counts[t] = 0;
  if (t < 8) accum[t] = 0.0f;
}

// ---------------- K1: per-sample feature inv-norm + bin histogram ----------------
__global__ void k_featnorm(const float* __restrict__ feats,
                           const int* __restrict__ bins,
                           float* __restrict__ inv_f, int* __restrict__ counts) {
  const int wave = threadIdx.x >> 5;
  const int lane = threadIdx.x & 31;
  const int sample = blockIdx.x * 8 + wave;
  if (sample >= BATCH) return;
  const float* row = feats + (size_t)sample * DIM;
  float s = 0.0f;
#pragma unroll
  for (int j = 0; j < DIM / 32; ++j) {
    float x = row[lane + 32 * j];
    s += x * x;
  }
  s = wave_sum(s);
  if (lane == 0) {
    inv_f[sample] = 1.0f / (sqrtf(s) + 1e-8f);
    atomicAdd(&counts[bins[sample]], 1);
  }
}

// ---------------- K2: per-(bin,pattern) centroid inv-norm ----------------
__global__ void k_centnorm(const float* __restrict__ cents, float* __restrict__ inv_c) {
  const int wave = threadIdx.x >> 5;
  const int lane = threadIdx.x & 31;
  const int row = blockIdx.x * 8 + wave;
  if (row >= NCLS) return;
  const float* r = cents + (size_t)row * DIM;
  float s = 0.0f;
#pragma unroll
  for (int j = 0; j < DIM / 32; ++j) {
    float x = r[lane + 32 * j];
    s += x * x;
  }
  s = wave_sum(s);
  if (lane == 0) inv_c[row] = 1.0f / (sqrtf(s) + 1e-8f);
}

// ---------------- K3: exclusive scan over 32 bin counts ----------------
__global__ void k_scan(const int* __restrict__ counts, int* __restrict__ offsets,
                       int* __restrict__ cursors) {
  if (threadIdx.x == 0) {
    int acc = 0;
    for (int i = 0; i < NUM_BINS; ++i) {
      offsets[i] = acc;
      cursors[i] = acc;
      acc += counts[i];
    }
  }
}

// ---------------- K4: scatter sample indices into bin-sorted order ----------------
__global__ void k_scatter(const int* __restrict__ bins, int* __restrict__ cursors,
                          int* __restrict__ order) {
  const int b = blockIdx.x * blockDim.x + threadIdx.x;
  if (b >= BATCH) return;
  const int pos = atomicAdd(&cursors[bins[b]], 1);
  order[pos] = b;
}

// ---------------- K5: WMMA sims tiles ----------------
// One wave per 16-sample tile of one bin. A = centroids[bin] (16xK), B = 16
// same-bin feature columns (Kx16), D accumulated with v_wmma_f32_16x16x4_f32.
__global__ __launch_bounds__(32) void k_wmma_sims(
    const float* __restrict__ feats, const float* __restrict__ cents,
    const float* __restrict__ inv_f, const float* __restrict__ inv_c,
    const int* __restrict__ order, const int* __restrict__ offsets,
    const int* __restrict__ counts, float* __restrict__ sims) {
  const int bin = blockIdx.x;
  const int tile = blockIdx.y;
  const int count = counts[bin];
  if (tile * 16 >= count) return;           // wave-uniform exit
  const int start = offsets[bin];
  const int lane = threadIdx.x;
  const int col = lane & 15;                // pattern row (A) / sample column (B,D)
  const int slot = tile * 16 + col;
  const bool valid = slot < count;
  const int sample = order[start + (valid ? slot : (count - 1))];
  const int kOff = (lane < 16) ? 0 : 2;     // ISA A/B frag layout for 16x16x4 f32
  const float* aPtr = cents + (size_t)(bin * NUM_PAT + col) * DIM + kOff;
  const float* bPtr = feats + (size_t)sample * DIM + kOff;

  v8f acc = {};
  for (int k0 = 0; k0 < DIM; k0 += 64) {
    if (k0 + 64 < DIM) {
      __builtin_prefetch(bPtr + k0 + 64, 0, 1);   // global_prefetch_b8
      __builtin_prefetch(aPtr + k0 + 64, 0, 1);
    }
#pragma unroll
    for (int k = k0; k < k0 + 64; k += 4) {
      v2f a = *(const v2f*)(aPtr + k);
      v2f b = *(const v2f*)(bPtr + k);
      // (neg_a, A, neg_b, B, c_mod, C, reuse_a, reuse_b)
      acc = __builtin_amdgcn_wmma_f32_16x16x4_f32(false, a, false, b,
                                                  (short)0, acc, false, false);
    }
  }

  // D layout: lanes 0-15 hold M=0..7 (vgpr r -> M=r), lanes 16-31 hold M=8..15
  const float fi = inv_f[sample];
  const int mBase = (lane < 16) ? 0 : 8;
  if (valid) {
#pragma unroll
    for (int r = 0; r < 8; ++r) {
      const int m = mBase + r;
      sims[(size_t)sample * NUM_PAT + m] = acc[r] * inv_c[bin * NUM_PAT + m] * fi;
    }
  }
}

// ---------------- K6: per-sample losses (one wave per sample) ----------------
__global__ void k_loss(const float* __restrict__ logits, const float* __restrict__ sims,
                       const int* __restrict__ bins, float* __restrict__ accum) {
  __shared__ float part[6];
  if (threadIdx.x < 6) part[threadIdx.x] = 0.0f;
  __syncthreads();

  const int wave = threadIdx.x >> 5;
  const int lane = threadIdx.x & 31;
  const int sample = blockIdx.x * 8 + wave;
  if (sample < BATCH) {
    const int bin = bins[sample];
    const float* lrow = logits + (size_t)sample * NCLS;

    // ---- full 512-logit max/argmax + logsumexp ----
    float l[NCLS / 32];
    float mv = -INFINITY; int mi = NCLS;
#pragma unroll
    for (int j = 0; j < NCLS / 32; ++j) {
      const int idx = lane + 32 * j;
      l[j] = lrow[idx];
      if (l[j] > mv || (l[j] == mv && idx < mi)) { mv = l[j]; mi = idx; }
    }
    wave_argmax(mv, mi);                       // mv = global max, mi = first argmax
    float se = 0.0f;
#pragma unroll
    for (int j = 0; j < NCLS / 32; ++j) se += __expf(l[j] - mv);
    se = wave_sum(se);
    const float logZ = mv + __logf(se);
    const int full_pred = mi;

    // ---- sims argmax (lanes 0-15 hold the 16 patterns) ----
    const float sim = (lane < NUM_PAT) ? sims[(size_t)sample * NUM_PAT + lane] : -INFINITY;
    float sv = sim; int si = (lane < NUM_PAT) ? lane : NUM_PAT;
    wave_argmax(sv, si);
    const float max_sim = sv;
    const int pattern_id = si;

    // ---- soft assignment probs: softmax(sims / 0.1) ----
    const float pe = (lane < NUM_PAT) ? __expf((sim - max_sim) * 10.0f) : 0.0f;
    const float psum = wave_sum(pe);
    const float prob = pe / psum;

    // ---- CE over the bin's logit slice ----
    const float ls = (lane < NUM_PAT) ? lrow[bin * NUM_PAT + lane] : -INFINITY;
    const float ce_b = wave_sum((lane < NUM_PAT) ? prob * (ls - logZ) : 0.0f);

    // ---- entropy of softmax over the slice ----
    const float smax = wave_max(ls);
    const float e = (lane < NUM_PAT) ? __expf(ls - smax) : 0.0f;
    const float ssum = wave_sum(e);
    const float pr = e / ssum;
    const float ent_b = wave_sum((lane < NUM_PAT) ? -pr * __logf(pr + 1e-8f) : 0.0f);

    if (lane == 0) {
      atomicAdd(&part[0], 1.0f - max_sim);                       // feature_sim
      atomicAdd(&part[1], -ce_b);                                // ce
      atomicAdd(&part[2], ent_b);                                // entropy
      atomicAdd(&part[3], (full_pred >> 4) == bin ? 1.0f : 0.0f);
      atomicAdd(&part[4], (full_pred & 15) == pattern_id ? 1.0f : 0.0f);
      atomicAdd(&part[5], full_pred == bin * NUM_PAT + pattern_id ? 1.0f : 0.0f);
    }
  }
  __syncthreads();
  if (threadIdx.x < 6) atomicAdd(&accum[threadIdx.x], part[threadIdx.x]);
}

// ---------------- K7: finalize the 7 scalar outputs ----------------
__global__ void k_final(const float* __restrict__ accum, float* __restrict__ out) {
  if (threadIdx.x == 0 && blockIdx.x == 0) {
    const float invB = 1.0f / (float)BATCH;
    const float feat = accum[0] * invB;
    const float ce   = accum[1] * invB;
    const float ent  = accum[2] * invB;
    const float div  = -ent;
    out[0] = 0.8f * feat + 0.2f * ce + 0.05f * div;  // (W_FEAT+W_ROSE)*feat + W_CE*ce + W_DIV*div
    out[1] = feat;
    out[2] = ce;
    out[3] = div;
    out[4] = accum[3] * invB;
    out[5] = accum[4] * invB;
    out[6] = accum[5] * invB;
  }
}

extern "C" void kernel_launch(void* const* d_in, const int* in_sizes, int n_in,
                              void* d_out, int out_size, void* d_ws, size_t ws_size,
                              hipStream_t stream) {
  const float* feats  = (const float*)d_in[0];   // [16384, 768]
  const float* logits = (const float*)d_in[1];   // [16384, 512]
  const float* cents  = (const float*)d_in[2];   // [32, 16, 768]
  const int*   bins   = (const int*)d_in[3];     // [16384]
  float* out = (float*)d_out;

  // workspace layout (floats then ints)
  float* wsF     = (float*)d_ws;
  float* inv_f   = wsF;                          // 16384
  float* inv_c   = wsF + BATCH;                  // 512
  float* simsBuf = inv_c + NCLS;                 // 16384*16
  float* accum   = simsBuf + (size_t)BATCH * NUM_PAT;  // 8
  int* wsI     = (int*)(accum + 8);
  int* counts  = wsI;                            // 32
  int* offsets = wsI + 32;                       // 32
  int* cursors = wsI + 64;                       // 32
  int* order   = wsI + 96;                       // 16384

  k_init<<<1, 64, 0, stream>>>(counts, accum);
  k_featnorm<<<BATCH / 8, 256, 0, stream>>>(feats, bins, inv_f, counts);
  k_centnorm<<<NCLS / 8, 256, 0, stream>>>(cents, inv_c);
  k_scan<<<1, 32, 0, stream>>>(counts, offsets, cursors);
  k_scatter<<<BATCH / 256, 256, 0, stream>>>(bins, cursors, order);
  k_wmma_sims<<<dim3(NUM_BINS, BATCH / 16), 32, 0, stream>>>(
      feats, cents, inv_f, inv_c, order, offsets, counts, simsBuf);
  k_loss<<<BATCH / 8, 256, 0, stream>>>(logits, simsBuf, bins, accum);
  k_final<<<1, 32, 0, stream>>>(accum, out);
}